// DualSignalLearning_33062658245234
// MI455X (gfx1250) — compile-verified
//
#include <hip/hip_runtime.h>
#include <math.h>

// ---------------- problem constants ----------------
#define Nn 64
#define Ff 4
#define Bb 32
#define Dd 1024
#define Rr 64

#define ETA_W   0.015f
#define ETA_S   0.003f
#define LAM_C   0.65f
#define LAM_R   0.35f
#define THETA0  0.5f
#define SPEC_MIN 0.3f
#define SPEC_MAX 4.0f
#define ADB1    0.9f
#define ADB2    0.999f
#define ADEPS   1e-8f
#define W_MAX   64.0f           // 2*sqrt(1024)
#define ALPHA_H 0.01f
#define BETA_H  0.05f

constexpr size_t NDR = (size_t)Nn * Dd * Rr;     // 4,194,304
constexpr size_t BD  = (size_t)Bb * Dd;          // 32,768

// output layout (floats): mse, A_new, B_new, S_new, rho_new, tensions
constexpr size_t OUT_A   = 1;
constexpr size_t OUT_B   = OUT_A + NDR;
constexpr size_t OUT_S   = OUT_B + NDR;
constexpr size_t OUT_RHO = OUT_S + (size_t)Nn * Dd;
constexpr size_t OUT_TEN = OUT_RHO + Nn;

// workspace layout (floats); vector-accessed blocks are 16B aligned
constexpr size_t WS_TNORM = 0;
constexpr size_t WS_L     = 1;                        // mean_b ||T_v[b,:]||
constexpr size_t WS_SPEC  = 64;                       // N
constexpr size_t WS_NORMS = 128;                      // N*F
constexpr size_t WS_TMEAN = 384;                      // D
constexpr size_t WS_C     = 1408;                     // N  (scan coefficients)
constexpr size_t WS_U     = 1536;                     // N*D
constexpr size_t WS_VM    = WS_U  + (size_t)Nn * Dd;  // N*D
constexpr size_t WS_TA    = WS_VM + (size_t)Nn * Dd;  // N*R
constexpr size_t WS_TB    = WS_TA + (size_t)Nn * Rr;  // N*R

typedef float v2f __attribute__((ext_vector_type(2)));
typedef float v4f __attribute__((ext_vector_type(4)));
typedef float v8f __attribute__((ext_vector_type(8)));

// ---------------- helpers ----------------

__device__ __forceinline__ float dot4(v4f a, v4f b) {
    return a.x * b.x + a.y * b.y + a.z * b.z + a.w * b.w;
}

__device__ __forceinline__ float block_reduce_sum_256(float v, float* red, int tid) {
    red[tid] = v;
    __syncthreads();
    for (int s = 128; s > 0; s >>= 1) {
        if (tid < s) red[tid] += red[tid + s];
        __syncthreads();
    }
    float r = red[0];
    __syncthreads();
    return r;
}

// t[0..63] = sum_d M[d*64 + r] * x[d], M is (1024 x 64) row-major, x in LDS.
// 8 waves: wave = 4*dh + rt ; rt selects 16 r's, dh selects 512-d half.
// WMMA f32 16x16x4: A-frag[m,k] = M[d0+k, r0+m], B-frag[k,n] = x[d0+k] (bcast),
// accumulator carries the K reduction across 128 chunks.
__device__ __forceinline__ void matvecT_wmma(const float* __restrict__ M,
                                             const float* xs, float* ts,
                                             float* part /*128*/, int tid) {
    const int lane  = tid & 31;
    const int wave  = tid >> 5;
    const int r0    = (wave & 3) << 4;   // 0,16,32,48
    const int dh    = wave >> 2;         // 0,1
    const int m     = lane & 15;
    const int hi    = lane >> 4;         // 0 or 1
    const int kbase = hi << 1;           // 0 or 2
    v8f c = {0.f, 0.f, 0.f, 0.f, 0.f, 0.f, 0.f, 0.f};
    const int dstart = dh << 9;          // 0 or 512
    for (int kc = 0; kc < 128; ++kc) {
        const int d0 = dstart + (kc << 2) + kbase;
        v2f a, b;
        a.x = M[(size_t)d0 * Rr + r0 + m];
        a.y = M[(size_t)(d0 + 1) * Rr + r0 + m];
        b.x = xs[d0];
        b.y = xs[d0 + 1];
        c = __builtin_amdgcn_wmma_f32_16x16x4_f32(false, a, false, b,
                                                  (short)0, c, false, false);
    }
    if (m == 0) {
        const int rbase = r0 + (hi << 3);
        #pragma unroll
        for (int j = 0; j < 8; ++j) part[(dh << 6) + rbase + j] = c[j];
    }
    __syncthreads();
    if (tid < 64) ts[tid] = part[tid] + part[64 + tid];
    __syncthreads();
}

// ---------------- kernels ----------------

// 1) mse, T_norm, and L = mean_b ||T_v[b,:]||.  one block of 1024 threads.
__global__ void k_mse(const float* __restrict__ Yh, const float* __restrict__ Ys,
                      float* __restrict__ out, float* __restrict__ ws) {
    __shared__ float red[1024];
    __shared__ float nbuf[32];
    const int tid = threadIdx.x;
    float s = 0.f;
    for (int e = tid; e < (int)BD; e += 1024) {
        float d = Ys[e] - Yh[e];
        s += d * d;
    }
    red[tid] = s;
    __syncthreads();
    for (int st = 512; st > 0; st >>= 1) {
        if (tid < st) red[tid] += red[tid + st];
        __syncthreads();
    }
    if (tid == 0) {
        float mse = red[0] / (float)BD;
        out[0] = mse;
        float tn = sqrtf(mse);
        ws[WS_TNORM] = fminf(fmaxf(tn, 0.f), 1.f);
    }
    __syncthreads();
    // phase 2: per-batch-row norms of T_v ; group of 32 lanes per row
    const int b = tid >> 5, j = tid & 31;
    float s2 = 0.f;
    for (int q = 0; q < 32; ++q) {
        const int e = b * Dd + j + q * 32;
        float d = (Ys[e] - Yh[e]) * (1.f / 32.f);
        s2 += d * d;
    }
    red[tid] = s2;
    __syncthreads();
    for (int st = 16; st > 0; st >>= 1) {
        if (j < st) red[tid] += red[tid + st];
        __syncthreads();
    }
    if (j == 0) nbuf[b] = sqrtf(red[tid]);
    __syncthreads();
    if (tid == 0) {
        float L = 0.f;
        for (int b2 = 0; b2 < Bb; ++b2) L += nbuf[b2];
        ws[WS_L] = L / (float)Bb;
    }
}

// 2) T_mean[d] = mean_b T_v = (sum_b (Ys-Yh)) / (B * sqrt(D))
__global__ void k_tmean(const float* __restrict__ Yh, const float* __restrict__ Ys,
                        float* __restrict__ ws) {
    const int d = blockIdx.x * 256 + threadIdx.x;
    float s = 0.f;
    for (int b = 0; b < Bb; ++b) s += Ys[b * Dd + d] - Yh[b * Dd + d];
    ws[WS_TMEAN + d] = s * (1.f / 1024.f);
}

// 3) spectral norm per node via power iteration; WMMA for B^T v and A^T u.
__global__ void k_spec(const float* __restrict__ A, const float* __restrict__ Bw,
                       float* __restrict__ ws) {
    __shared__ float v_lds[Dd];
    __shared__ float u_lds[Dd];
    __shared__ __align__(16) float t_lds[64];
    __shared__ float part[128];
    __shared__ float red[256];
    const int n = blockIdx.x, tid = threadIdx.x;
    const float* An = A  + (size_t)n * Dd * Rr;
    const float* Bn = Bw + (size_t)n * Dd * Rr;
    for (int q = 0; q < 4; ++q) v_lds[tid + q * 256] = 1.0f / 32.0f;
    __syncthreads();
    for (int it = 0; it < 12; ++it) {
        matvecT_wmma(Bn, v_lds, t_lds, part, tid);          // t = B^T v
        for (int q = 0; q < 4; ++q) {                        // u = A t
            const int d = tid + q * 256;
            const v4f* Arow = (const v4f*)(An + (size_t)d * Rr);
            float acc = 0.f;
            for (int rr = 0; rr < 16; ++rr)
                acc += dot4(Arow[rr], *(const v4f*)(t_lds + rr * 4));
            u_lds[d] = acc;
        }
        __syncthreads();
        matvecT_wmma(An, u_lds, t_lds, part, tid);          // s = A^T u
        float w4[4];
        float ss = 0.f;
        for (int q = 0; q < 4; ++q) {                        // w = B s
            const int d = tid + q * 256;
            const v4f* Brow = (const v4f*)(Bn + (size_t)d * Rr);
            float acc = 0.f;
            for (int rr = 0; rr < 16; ++rr)
                acc += dot4(Brow[rr], *(const v4f*)(t_lds + rr * 4));
            w4[q] = acc;
            ss += acc * acc;
        }
        float nrm = sqrtf(block_reduce_sum_256(ss, red, tid));
        float inv = 1.0f / (nrm + 1e-9f);
        for (int q = 0; q < 4; ++q) v_lds[tid + q * 256] = w4[q] * inv;
        __syncthreads();
    }
    matvecT_wmma(Bn, v_lds, t_lds, part, tid);              // t = B^T v
    float ss = 0.f;
    for (int q = 0; q < 4; ++q) {                            // || A t ||
        const int d = tid + q * 256;
        const v4f* Arow = (const v4f*)(An + (size_t)d * Rr);
        float acc = 0.f;
        for (int rr = 0; rr < 16; ++rr)
            acc += dot4(Arow[rr], *(const v4f*)(t_lds + rr * 4));
        ss += acc * acc;
    }
    float spec = sqrtf(block_reduce_sum_256(ss, red, tid));
    if (tid == 0) ws[WS_SPEC + n] = fminf(fmaxf(spec, SPEC_MIN), SPEC_MAX);
}

// 4) blame norms: norms[i,f] = mask * mean_b ||contribs[i,f,b,:]||
__global__ void k_blame(const float* __restrict__ contribs, float* __restrict__ ws) {
    __shared__ float sq[256];
    __shared__ float nb[32];
    const int i = blockIdx.x >> 2, f = blockIdx.x & 3;
    const int tid = threadIdx.x;
    const int b = tid >> 3, j = tid & 7;
    const float mask = ((i - 1 - f) >= 0) ? 1.f : 0.f;
    const size_t base = (((size_t)i * Ff + f) * Bb + b) * Dd + (size_t)j * 128;
    const v4f* p = (const v4f*)(contribs + base);
    float s = 0.f;
    for (int q = 0; q < 32; ++q) {
        v4f x = p[q];
        s += dot4(x, x);
    }
    sq[tid] = s;
    __syncthreads();
    if (j == 0) {
        float t = 0.f;
        for (int k = 0; k < 8; ++k) t += sq[(b << 3) + k];
        nb[b] = sqrtf(t);
    }
    __syncthreads();
    if (tid == 0) {
        float t = 0.f;
        for (int b2 = 0; b2 < Bb; ++b2) t += nb[b2];
        ws[WS_NORMS + i * Ff + f] = mask * (t / (float)Bb);
    }
}

// 5) scan coefficients: tensions[n] == c[n] * T_v  (linear scan w/ scalar weights;
//    only node N-1 has a nonzero initial tension, so the whole retrograde scan
//    collapses to this 64-element scalar recurrence).
__global__ void k_coeff(float* __restrict__ ws) {
    __shared__ float c[Nn];
    const int tid = threadIdx.x;   // 64 threads
    c[tid] = (tid == Nn - 1) ? 1.f : 0.f;
    __syncthreads();
    if (tid == 0) {
        for (int i = Nn - 1; i >= 1; --i) {
            const float n0 = ws[WS_NORMS + i * Ff + 0];
            const float n1 = ws[WS_NORMS + i * Ff + 1];
            const float n2 = ws[WS_NORMS + i * Ff + 2];
            const float n3 = ws[WS_NORMS + i * Ff + 3];
            const float inv = 1.f / (n0 + n1 + n2 + n3 + 1e-9f);
            const float sp = ws[WS_SPEC + i];
            const float ci = c[i];
            if (i - 1 >= 0) c[i - 1] += ci * n0 * inv * sp;
            if (i - 2 >= 0) c[i - 2] += ci * n1 * inv * sp;
            if (i - 3 >= 0) c[i - 3] += ci * n2 * inv * sp;
            if (i - 4 >= 0) c[i - 4] += ci * n3 * inv * sp;
        }
    }
    __syncthreads();
    ws[WS_C + tid] = c[tid];
}

// 6) materialize tensions[n,b,d] = c[n] * T_v[b,d]
__global__ void k_tfill(const float* __restrict__ Yh, const float* __restrict__ Ys,
                        const float* __restrict__ ws, float* __restrict__ out) {
    const size_t e4 = ((size_t)blockIdx.x * 256 + threadIdx.x) * 4;  // [0, N*B*D)
    const int n = (int)(e4 >> 15);
    const size_t j = e4 & (BD - 1);
    const float cn = ws[WS_C + n] * (1.f / 32.f);
    const v4f ys = *(const v4f*)(Ys + j);
    const v4f yh = *(const v4f*)(Yh + j);
    const v4f tv = (ys - yh) * cn;
    out[OUT_TEN + e4 + 0] = tv.x;
    out[OUT_TEN + e4 + 1] = tv.y;
    out[OUT_TEN + e4 + 2] = tv.z;
    out[OUT_TEN + e4 + 3] = tv.w;
}

// 7) per-node update vectors u (clipped) and v_m.
//    mean_b T_n[d] == c[n] * T_mean[d], so no tension reads needed.
__global__ void k_nodevec(const float* __restrict__ Vin, const float* __restrict__ Vout,
                          const float* __restrict__ Vw, const float* __restrict__ goodness,
                          const float* __restrict__ ws_c, float* __restrict__ ws) {
    __shared__ float red[256];
    const int n = blockIdx.x, tid = threadIdx.x;
    const size_t base = (size_t)n * BD;
    const v4f* Vin4  = (const v4f*)(Vin + base);
    const v4f* Vout4 = (const v4f*)(Vout + base);
    const v4f* Vw4   = (const v4f*)(Vw + base);
    // theta = THETA0 * (mean_b sum_d Vin^2 + 1e-9)
    float s = 0.f;
    for (int p = 0; p < 32; ++p) {
        v4f x = Vin4[p * 256 + tid];
        s += dot4(x, x);
    }
    const float tsum = block_reduce_sum_256(s, red, tid);
    const float theta = THETA0 * (tsum / (float)Bb + 1e-9f);
    const float dg = goodness[n] - theta;
    const float cn = ws_c[WS_C + n];

    v4f sfp = {0.f, 0.f, 0.f, 0.f};
    v4f svofp = {0.f, 0.f, 0.f, 0.f};
    v4f svin = {0.f, 0.f, 0.f, 0.f};
    for (int b = 0; b < Bb; ++b) {
        const int idx = b * 256 + tid;
        const v4f vw = Vw4[idx];
        v4f fp;
        #pragma unroll
        for (int k = 0; k < 4; ++k) {
            float th = tanhf(vw[k]);
            fp[k] = 1.f - th * th;
        }
        sfp   += fp;
        svofp += Vout4[idx] * fp;
        svin  += Vin4[idx];
    }
    const v4f tm = *(const v4f*)(ws_c + WS_TMEAN + tid * 4);
    const v4f u_c = (dg / (float)Bb) * svofp;
    const v4f u_r = (cn * tm) * (sfp * (1.f / (float)Bb));
    const v4f vm = svin * (1.f / (float)Bb);
    const v4f u = LAM_C * u_c + LAM_R * u_r;
    const float un = block_reduce_sum_256(dot4(u, u), red, tid);
    const float vn = block_reduce_sum_256(dot4(vm, vm), red, tid);
    const float gnorm = sqrtf(un) * sqrtf(vn);
    const float sc = fminf(1.f, 5.f / (gnorm + 1e-12f));
    *(v4f*)(ws + WS_U  + (size_t)n * Dd + tid * 4) = u * sc;
    *(v4f*)(ws + WS_VM + (size_t)n * Dd + tid * 4) = vm;
}

// 8) t_A = B^T v_m ; t_B = A^T u  (WMMA matvecs)
__global__ void k_tatb(const float* __restrict__ A, const float* __restrict__ Bw,
                       float* __restrict__ ws) {
    __shared__ __align__(16) float x_lds[Dd];
    __shared__ __align__(16) float t_lds[64];
    __shared__ float part[128];
    const int n = blockIdx.x, tid = threadIdx.x;
    const float* An = A  + (size_t)n * Dd * Rr;
    const float* Bn = Bw + (size_t)n * Dd * Rr;
    *(v4f*)(x_lds + tid * 4) = *(const v4f*)(ws + WS_VM + (size_t)n * Dd + tid * 4);
    __syncthreads();
    matvecT_wmma(Bn, x_lds, t_lds, part, tid);
    if (tid < 64) ws[WS_TA + (size_t)n * Rr + tid] = t_lds[tid];
    __syncthreads();
    *(v4f*)(x_lds + tid * 4) = *(const v4f*)(ws + WS_U + (size_t)n * Dd + tid * 4);
    __syncthreads();
    matvecT_wmma(An, x_lds, t_lds, part, tid);
    if (tid < 64) ws[WS_TB + (size_t)n * Rr + tid] = t_lds[tid];
}

__device__ __forceinline__ float adam_step(float W, float m, float v, float g) {
    const float m2 = ADB1 * m + (1.f - ADB1) * g;
    const float v2 = ADB2 * v + (1.f - ADB2) * g * g;
    const float mh = m2 / (1.f - ADB1);
    const float vh = v2 / (1.f - ADB2);
    float Wn = W - ETA_W * mh / (sqrtf(vh) + ADEPS);
    return fminf(fmaxf(Wn, -W_MAX), W_MAX);
}

// 9) Adam update of A and B (elementwise stream; node 0 passthrough)
__global__ void k_adam(const float* __restrict__ A, const float* __restrict__ Bw,
                       const float* __restrict__ mA, const float* __restrict__ vA,
                       const float* __restrict__ mB, const float* __restrict__ vB,
                       const float* __restrict__ ws, float* __restrict__ out) {
    const size_t e4 = ((size_t)blockIdx.x * 256 + threadIdx.x) * 4;  // [0, N*D*R)
    const int n = (int)(e4 >> 16);
    const int d = (int)((e4 >> 6) & (Dd - 1));
    const int r = (int)(e4 & (Rr - 1));
    const float u  = ws[WS_U  + (size_t)n * Dd + d];
    const float vm = ws[WS_VM + (size_t)n * Dd + d];
    const v4f ta = *(const v4f*)(ws + WS_TA + (size_t)n * Rr + r);
    const v4f tb = *(const v4f*)(ws + WS_TB + (size_t)n * Rr + r);
    const v4f a  = *(const v4f*)(A  + e4);
    const v4f b  = *(const v4f*)(Bw + e4);
    const v4f ma = *(const v4f*)(mA + e4);
    const v4f va = *(const v4f*)(vA + e4);
    const v4f mb = *(const v4f*)(mB + e4);
    const v4f vb = *(const v4f*)(vB + e4);
    #pragma unroll
    for (int k = 0; k < 4; ++k) {
        float aNew = adam_step(a[k], ma[k], va[k], u * ta[k]);
        float bNew = adam_step(b[k], mb[k], vb[k], vm * tb[k]);
        if (n == 0) { aNew = a[k]; bNew = b[k]; }
        out[OUT_A + e4 + k] = aNew;
        out[OUT_B + e4 + k] = bNew;
    }
}

// 10) health / rho : T_local[n] = |c[n]| * L
__global__ void k_rho(const float* __restrict__ rho, const float* __restrict__ ws,
                      float* __restrict__ out) {
    const int n = threadIdx.x;  // 64 threads
    const float Tl = fabsf(ws[WS_C + n]) * ws[WS_L];
    const float Tn = ws[WS_TNORM];
    const float wb = (Tn > 0.f) ? fminf(1.f, Tl / (Tn + 1e-9f)) : 1.f;
    float delta = ALPHA_H * (1.f - Tn) - BETA_H * (1.f + Tn * Tn) * Tn * wb;
    delta = fminf(fmaxf(delta, -0.1f), 0.1f);
    out[OUT_RHO + n] = fminf(fmaxf(rho[n] + delta, -5.f), 10.f);
}

// 11) signature update
__global__ void k_sig(const float* __restrict__ S, const float* __restrict__ r_dot,
                      const float* __restrict__ ws, float* __restrict__ out) {
    __shared__ float red[256];
    const int n = blockIdx.x, tid = threadIdx.x;
    const float rd = r_dot[n];
    const v4f sv = *(const v4f*)(S + (size_t)n * Dd + tid * 4);
    const v4f tm = *(const v4f*)(ws + WS_TMEAN + tid * 4);
    const v4f s2 = sv + (ETA_S * rd) * tm;
    const float nrm = sqrtf(block_reduce_sum_256(dot4(s2, s2), red, tid));
    const float inv = (nrm > 1e-9f) ? (1.f / nrm) : 1.f;
    #pragma unroll
    for (int k = 0; k < 4; ++k) {
        const float val = (n == 0) ? sv[k] : s2[k] * inv;
        out[OUT_S + (size_t)n * Dd + tid * 4 + k] = val;
    }
}

// ---------------- launch ----------------
extern "C" void kernel_launch(void* const* d_in, const int* in_sizes, int n_in,
                              void* d_out, int out_size, void* d_ws, size_t ws_size,
                              hipStream_t stream) {
    const float* Yh       = (const float*)d_in[0];
    const float* Ys       = (const float*)d_in[1];
    const float* contribs = (const float*)d_in[2];
    const float* Vin      = (const float*)d_in[3];
    const float* Vout     = (const float*)d_in[4];
    const float* Vw       = (const float*)d_in[5];
    const float* goodness = (const float*)d_in[6];
    const float* A        = (const float*)d_in[7];
    const float* Bw       = (const float*)d_in[8];
    const float* mA       = (const float*)d_in[9];
    const float* vA       = (const float*)d_in[10];
    const float* mB       = (const float*)d_in[11];
    const float* vB       = (const float*)d_in[12];
    const float* S        = (const float*)d_in[13];
    const float* rho      = (const float*)d_in[14];
    const float* r_dot    = (const float*)d_in[15];
    // src_ids / src_mask (d_in[16], d_in[17]) encode the deterministic chain
    // src=max(i-1-f,0), mask=(i-1-f)>=0 -- handled analytically.
    float* out = (float*)d_out;
    float* ws  = (float*)d_ws;

    k_mse   <<<1, 1024, 0, stream>>>(Yh, Ys, out, ws);
    k_tmean <<<Dd / 256, 256, 0, stream>>>(Yh, Ys, ws);
    k_spec  <<<Nn, 256, 0, stream>>>(A, Bw, ws);
    k_blame <<<Nn * Ff, 256, 0, stream>>>(contribs, ws);
    k_coeff <<<1, Nn, 0, stream>>>(ws);
    k_tfill <<<(unsigned)((size_t)Nn * BD / 1024), 256, 0, stream>>>(Yh, Ys, ws, out);
    k_nodevec<<<Nn, 256, 0, stream>>>(Vin, Vout, Vw, goodness, ws, ws);
    k_tatb  <<<Nn, 256, 0, stream>>>(A, Bw, ws);
    k_adam  <<<(unsigned)(NDR / 1024), 256, 0, stream>>>(A, Bw, mA, vA, mB, vB, ws, out);
    k_rho   <<<1, Nn, 0, stream>>>(rho, ws, out);
    k_sig   <<<Nn, 256, 0, stream>>>(S, r_dot, ws, out);
}